// SplitLayer_62603443306880
// MI455X (gfx1250) — compile-verified
//
#include <hip/hip_runtime.h>
#include <hip/hip_bf16.h>
#include <stdint.h>

// Problem constants (match reference)
#define BB 512
#define LL 8192
#define NN 128
#define SS 256
#define DLM 5
#define NT 256            // threads per block (8 wave32 waves)
#define CHUNK (LL / NT)   // 32 tokens per thread

typedef unsigned int v4u __attribute__((ext_vector_type(4)));
typedef int          v8i __attribute__((ext_vector_type(8)));
typedef int          v4i __attribute__((ext_vector_type(4)));

__global__ __launch_bounds__(NT)
void split_rows_kernel(const int* __restrict__ x,
                       float* __restrict__ otp,      // [B, N, S]
                       float* __restrict__ len_doc,  // [B]
                       float* __restrict__ mask)     // [B, N, S]
{
    __shared__ __align__(16) int s_x[LL];   // 32 KB row staging (TDM target)
    __shared__ int s_part[NT];              // scan partials
    __shared__ int s_starts[NN + 1];        // chunk start positions in rpad
    __shared__ int s_len[NN];               // len_sent per chunk

    const int b    = blockIdx.x;
    const int tid  = threadIdx.x;
    const int lane = tid & 31;
    const int wave = tid >> 5;

    // ---- init starts while wave 0 kicks off the TDM DMA -------------------
    if (tid <= NN) s_starts[tid] = LL + 1;  // unused chunks -> start = L+1
    if (tid == 0)  s_starts[0] = 0;

#if __has_builtin(__builtin_amdgcn_tensor_load_to_lds) && __has_builtin(__builtin_amdgcn_s_wait_tensorcnt)
    if (tid < 32) {  // only wave 0 reaches the TDM issue (EXEC is ignored by TDM)
        const uint64_t ga  = (uint64_t)(uintptr_t)(x + (size_t)b * LL);
        const uint32_t lds = (uint32_t)(uintptr_t)&s_x[0];  // low 32b of flat = LDS offset
        // D# group0: count=1 | lds_addr | global_addr[56:0] | type=2
        v4u g0;
        g0.x = 1u;                                   // count=1, user descriptor
        g0.y = lds;                                  // lds_addr (bytes)
        g0.z = (uint32_t)ga;                         // global_addr[31:0]
        g0.w = ((uint32_t)(ga >> 32) & 0x01FFFFFFu)  // global_addr[56:32]
             | 0x80000000u;                          // type=2 ("image")
        // D# group1: data_size=4B; tensor 8192x1; tile 8192x1; stride0=8192
        v8i g1;
        g1[0] = 0x00020000;          // workgroup_mask=0, data_size=2 (4 bytes)
        g1[1] = (int)(LL << 16);     // tensor_dim0[15:0] << 16  (8192)
        g1[2] = 0x00010000;          // tensor_dim0[31:16]=0, tensor_dim1[15:0]=1
        g1[3] = (int)(LL << 16);     // tensor_dim1[31:16]=0, tile_dim0=8192
        g1[4] = 0x00000001;          // tile_dim1=1, tile_dim2=0
        g1[5] = LL;                  // tensor_dim0_stride[31:0] = 8192
        g1[6] = 0;                   // stride0 hi / stride1 lo
        g1[7] = 0;                   // stride1 hi
        v4i gz4 = {0, 0, 0, 0};      // groups 2/3 unused (2-D tensor)
        v8i gz8 = {0, 0, 0, 0, 0, 0, 0, 0};
        __builtin_amdgcn_tensor_load_to_lds(g0, g1, gz4, gz4, gz8, 0);
        __builtin_amdgcn_s_wait_tensorcnt(0);
    }
#else
    // Fallback: cooperative coalesced load (one int4 per lane per iter)
    for (int i = tid * 4; i < LL; i += NT * 4) {
        *(int4*)&s_x[i] = *(const int4*)&x[(size_t)b * LL + i];
    }
#endif
    __syncthreads();

    // ---- pass 1: per-thread delimiter count ------------------------------
    const int base = tid * CHUNK;
    int cnt = 0;
#pragma unroll
    for (int i = 0; i < CHUNK; i += 4) {
        int4 v = *(const int4*)&s_x[base + i];
        cnt += (v.x == DLM) + (v.y == DLM) + (v.z == DLM) + (v.w == DLM);
    }
    s_part[tid] = cnt;
    __syncthreads();

    // ---- block-wide inclusive scan (Hillis-Steele, 8 steps) --------------
    int run = cnt;
#pragma unroll
    for (int off = 1; off < NT; off <<= 1) {
        int add = (tid >= off) ? s_part[tid - off] : 0;
        __syncthreads();
        run += add;
        s_part[tid] = run;
        __syncthreads();
    }
    const int excl = run - cnt;  // delimiters strictly before this thread's span

    // ---- pass 2: record chunk starts (delimiter d opens chunk d+1) -------
    int cum = excl;
#pragma unroll
    for (int i = 0; i < CHUNK; i += 4) {
        int4 v = *(const int4*)&s_x[base + i];
        int a0 = v.x, a1 = v.y, a2 = v.z, a3 = v.w;
        if (a0 == DLM) { if (cum < NN - 1) s_starts[cum + 1] = base + i + 1; cum++; }
        if (a1 == DLM) { if (cum < NN - 1) s_starts[cum + 1] = base + i + 2; cum++; }
        if (a2 == DLM) { if (cum < NN - 1) s_starts[cum + 1] = base + i + 3; cum++; }
        if (a3 == DLM) { if (cum < NN - 1) s_starts[cum + 1] = base + i + 4; cum++; }
    }
    __syncthreads();

    // ---- pass 3: gather otp rows (coalesced) + len_sent via ballot -------
    float* __restrict__ otp_row = otp + (size_t)b * NN * SS;
    for (int k = wave; k < NN; k += (NT / 32)) {
        const int st = s_starts[k];
        const int sz = s_starts[k + 1] - st;  // chunk size (0 if unused)
        int nz = 0;
#pragma unroll
        for (int s0 = 0; s0 < SS; s0 += 32) {
            const int s  = s0 + lane;
            const int j  = st + s;
            const int jc = (j < LL) ? j : 0;         // clamp for safe LDS read
            const int tok = s_x[jc];
            const bool in = (sz != 1) & (s < sz);    // size-1 chunk -> PAD
            const int val = in ? ((j == LL) ? DLM : tok) : 0;  // rpad trailing delim
            otp_row[k * SS + s] = (float)val;
            unsigned bal = __builtin_amdgcn_ballot_w32(val != 0);
            nz += __builtin_popcount(bal);
        }
        if (lane == 0) s_len[k] = nz;
    }
    __syncthreads();

    // ---- pass 4: mask rows (coalesced) + len_doc -------------------------
    float* __restrict__ mask_row = mask + (size_t)b * NN * SS;
    for (int k = wave; k < NN; k += (NT / 32)) {
        const int len = s_len[k];
#pragma unroll
        for (int s0 = 0; s0 < SS; s0 += 32) {
            const int s = s0 + lane;
            mask_row[k * SS + s] = (s < len) ? 1.0f : 0.0f;
        }
    }
    if (tid == 0) {
        int c = 0;
#pragma unroll 4
        for (int k = 0; k < NN; ++k) c += (s_len[k] != 0);
        len_doc[b] = (float)c;
    }
}

extern "C" void kernel_launch(void* const* d_in, const int* in_sizes, int n_in,
                              void* d_out, int out_size, void* d_ws, size_t ws_size,
                              hipStream_t stream) {
    (void)in_sizes; (void)n_in; (void)out_size; (void)d_ws; (void)ws_size;
    const int* x = (const int*)d_in[0];
    float* out     = (float*)d_out;
    float* otp     = out;                                   // B*N*S
    float* len_doc = out + (size_t)BB * NN * SS;            // B
    float* mask    = len_doc + BB;                          // B*N*S
    split_rows_kernel<<<dim3(BB), dim3(NT), 0, stream>>>(x, otp, len_doc, mask);
}